// FocalLoss_59622736003393
// MI455X (gfx1250) — compile-verified
//
#include <hip/hip_runtime.h>
#include <stdint.h>

#define BLOCK 256
#define B_ 8
#define A_ 200000
#define M_ 50
#define C_ 20
#define NCHUNK ((A_ + BLOCK - 1) / BLOCK)      // 782 anchor chunks
#define CPB 8                                   // chunks per block (double-buffered TDM pipeline)
#define NBX ((NCHUNK + CPB - 1) / CPB)          // 98 blocks per batch

typedef unsigned int u32x4 __attribute__((ext_vector_type(4)));
typedef int          i32x4 __attribute__((ext_vector_type(4)));
typedef int          i32x8 __attribute__((ext_vector_type(8)));

#if defined(__HIP_DEVICE_COMPILE__) && __has_builtin(__builtin_amdgcn_tensor_load_to_lds) && __has_builtin(__builtin_amdgcn_s_wait_tensorcnt)
#define USE_TDM 1
#else
#define USE_TDM 0
#endif

#if USE_TDM
// Issue one TDM descriptor: DMA a [rows x 20] f32 tile (row-major, stride 20)
// from global memory into LDS at lds_off. Uniform args -> SGPR descriptor.
__device__ __forceinline__ void tdm_issue(const float* gptr, uint32_t lds_off, int rows) {
    const uint64_t ga = (uint64_t)(uintptr_t)gptr;

    // ---- D# group 0 (128b): count=1 user descriptor, lds_addr, global_addr[56:0], type=2
    u32x4 g0;
    g0.x = 1u;                                            // count=1, is_restore=0, gather off
    g0.y = lds_off;                                       // lds_addr (bytes)
    g0.z = (uint32_t)(ga & 0xFFFFFFFFu);                  // global_addr[31:0]
    g0.w = (uint32_t)((ga >> 32) & 0x1FFFFFFu)            // global_addr[56:32]
         | (2u << 30);                                    // type=2 ("image")

    // ---- D# group 1 (256b): 2D tile, data_size=4B
    i32x8 g1;
    g1[0] = (int)(2u << 16);                              // wg_mask=0, data_size=2 (4B)
    g1[1] = (int)((uint32_t)C_ << 16);                    // tensor_dim0 = 20 (bits[63:48])
    g1[2] = (int)((uint32_t)rows << 16);                  // tensor_dim0 hi=0 | tensor_dim1 lo16
    g1[3] = (int)((uint32_t)C_ << 16);                    // tensor_dim1 hi=0 | tile_dim0 = 20
    g1[4] = (int)(uint32_t)rows;                          // tile_dim1 = rows | tile_dim2 = 0
    g1[5] = C_;                                           // tensor_dim0_stride = 20 elems
    g1[6] = 0;
    g1[7] = 0;

    i32x4 z4 = {0, 0, 0, 0};                              // groups 2/3: unused (<=2D tensor)
    i32x8 z8 = {0, 0, 0, 0, 0, 0, 0, 0};
    __builtin_amdgcn_tensor_load_to_lds(g0, g1, z4, z4, z8, 0);
}
#endif

// Kernel 1: each block handles CPB=8 chunks of 256 anchors for one batch.
// Two 20 KB LDS tiles are ping-ponged: the TDM streams chunk i+1 while the
// waves compute IoU assignment + focal loss on chunk i. Per-block partial
// loss sum and positive count are tree-reduced (fixed order, deterministic).
__global__ void focal_main(const float* __restrict__ cls,
                           const float* __restrict__ anch,
                           const float* __restrict__ ann,
                           float* __restrict__ loss_part,
                           float* __restrict__ pos_part) {
    __shared__ __align__(16) float tile[2][BLOCK * C_];  // 2 x 20 KB TDM destinations
    __shared__ float s_ann[M_ * 3];
    __shared__ float s_red[BLOCK];
    __shared__ float s_cnt[BLOCK];

    const int tid = threadIdx.x;
    const int b   = blockIdx.y;
    const int c0  = blockIdx.x * CPB;
    const int nch = (NCHUNK - c0 < CPB) ? (NCHUNK - c0) : CPB;
    const float* cls_b = cls + (size_t)b * A_ * C_;

    // Stage this batch's annotations (50x3 f32 = 600 B) into LDS.
    if (tid < M_ * 3) s_ann[tid] = ann[b * (M_ * 3) + tid];

#if USE_TDM
    if (tid < 32) {  // wave 0 drives the TDM pipeline (EXEC ignored; per-wave TENSORcnt)
        const int a0 = c0 * BLOCK;
        const int r0 = (A_ - a0 < BLOCK) ? (A_ - a0) : BLOCK;
        tdm_issue(cls_b + (size_t)a0 * C_, (uint32_t)(uintptr_t)(void*)&tile[0][0], r0);
    }
#endif

    float lsum = 0.0f, posf = 0.0f;

    for (int ci = 0; ci < nch; ++ci) {
        const int chunk = c0 + ci;
        const int a0    = chunk * BLOCK;
        const int rows  = (A_ - a0 < BLOCK) ? (A_ - a0) : BLOCK;
        const int buf   = ci & 1;

#if USE_TDM
        if (tid < 32) {
            if (ci + 1 < nch) {   // prefetch next chunk into the other tile
                const int a1 = (chunk + 1) * BLOCK;
                const int r1 = (A_ - a1 < BLOCK) ? (A_ - a1) : BLOCK;
                tdm_issue(cls_b + (size_t)a1 * C_,
                          (uint32_t)(uintptr_t)(void*)&tile[buf ^ 1][0], r1);
                __builtin_amdgcn_s_wait_tensorcnt(1);   // oldest (current tile) done
            } else {
                __builtin_amdgcn_s_wait_tensorcnt(0);   // drain
            }
        }
#else
        for (int i = tid; i < rows * C_; i += BLOCK)
            tile[buf][i] = cls_b[(size_t)a0 * C_ + i];
#endif
        __syncthreads();   // tile[buf] (and, first pass, s_ann) visible to all waves

        if (tid < rows) {
            const int a = a0 + tid;
            const float ax0 = anch[2 * a];
            const float ax1 = anch[2 * a + 1];
            const float aw  = ax1 - ax0;

            // IoU max / argmax over M=50 (strict > == jnp.argmax first occurrence)
            float best = -1.0f, best_lab = 0.0f;
            for (int m = 0; m < M_; ++m) {
                const float b0 = s_ann[3 * m];
                const float b1 = s_ann[3 * m + 1];
                const float lb = s_ann[3 * m + 2];
                float iw = fminf(ax1, b1) - fmaxf(ax0, b0);
                iw = fmaxf(iw, 0.0f);
                const float ua  = fmaxf(aw + (b1 - b0) - iw, 1e-8f);
                const float iou = iw / ua;
                if (iou > best) { best = iou; best_lab = lb; }
            }

            const bool pos = (best >= 0.5f);
            const bool neg = (best < 0.4f);
            if (pos) posf += 1.0f;

            if (pos || neg) {
                const int k = pos ? (int)best_lab : -1;   // one-hot class for positives
                const float4* row = (const float4*)(&tile[buf][0] + tid * C_);  // 80 B, 16 B aligned
#pragma unroll
                for (int q = 0; q < 5; ++q) {
                    const float4 v = row[q];
                    const float pv[4] = {v.x, v.y, v.z, v.w};
#pragma unroll
                    for (int j = 0; j < 4; ++j) {
                        const int   c = q * 4 + j;
                        const float p = fminf(fmaxf(pv[j], 1e-4f), 1.0f - 1e-4f);
                        if (c == k) {
                            // target==1: alpha * (1-p)^2 * -log(p)
                            lsum += 0.25f * (1.0f - p) * (1.0f - p) * (-__logf(p));
                        } else {
                            // target==0: (1-alpha) * p^2 * -log(1-p)
                            lsum += 0.75f * p * p * (-__logf(1.0f - p));
                        }
                    }
                }
            }
        }

        __syncthreads();   // all reads of tile[buf] done before TDM overwrites it (iter ci+2)
    }

    // Deterministic block tree reduction of loss sum and positive count.
    s_red[tid] = lsum;
    s_cnt[tid] = posf;
    __syncthreads();
#pragma unroll
    for (int off = BLOCK / 2; off > 0; off >>= 1) {
        if (tid < off) {
            s_red[tid] += s_red[tid + off];
            s_cnt[tid] += s_cnt[tid + off];
        }
        __syncthreads();
    }
    if (tid == 0) {
        loss_part[(size_t)b * NBX + blockIdx.x] = s_red[0];
        pos_part [(size_t)b * NBX + blockIdx.x] = s_cnt[0];
    }
}

// Kernel 2: single block; deterministic reduction of per-block partials,
// per-image normalization by max(num_pos,1), mean over B.
__global__ void focal_finalize(const float* __restrict__ loss_part,
                               const float* __restrict__ pos_part,
                               float* __restrict__ out) {
    __shared__ float s[256];
    const int tid = threadIdx.x;
    float total = 0.0f;
    for (int b = 0; b < B_; ++b) {
        float ls = 0.0f, ps = 0.0f;
        for (int i = tid; i < NBX; i += 256) {
            ls += loss_part[(size_t)b * NBX + i];
            ps += pos_part [(size_t)b * NBX + i];
        }
        s[tid] = ls; __syncthreads();
        for (int off = 128; off > 0; off >>= 1) {
            if (tid < off) s[tid] += s[tid + off];
            __syncthreads();
        }
        const float lsum = s[0]; __syncthreads();
        s[tid] = ps; __syncthreads();
        for (int off = 128; off > 0; off >>= 1) {
            if (tid < off) s[tid] += s[tid + off];
            __syncthreads();
        }
        const float psum = s[0]; __syncthreads();
        if (tid == 0) total += lsum / fmaxf(psum, 1.0f);
    }
    if (tid == 0) out[0] = total * (1.0f / (float)B_);
}

extern "C" void kernel_launch(void* const* d_in, const int* in_sizes, int n_in,
                              void* d_out, int out_size, void* d_ws, size_t ws_size,
                              hipStream_t stream) {
    (void)in_sizes; (void)n_in; (void)out_size; (void)ws_size;
    const float* cls  = (const float*)d_in[0];  // [B, A, C] f32
    const float* anch = (const float*)d_in[1];  // [1, A, 2] f32
    const float* ann  = (const float*)d_in[2];  // [B, M, 3] f32
    float* out = (float*)d_out;

    float* loss_part = (float*)d_ws;                   // [B, NBX]
    float* pos_part  = loss_part + (size_t)B_ * NBX;   // [B, NBX]  (~6.3 KB total)

    dim3 grid(NBX, B_, 1);
    focal_main<<<grid, BLOCK, 0, stream>>>(cls, anch, ann, loss_part, pos_part);
    focal_finalize<<<1, 256, 0, stream>>>(loss_part, pos_part, out);
}